// MacUnit_90469191122998
// MI455X (gfx1250) — compile-verified
//
#include <hip/hip_runtime.h>
#include <hip/hip_bf16.h>
#include <stdint.h>

#define IN_CHANNELS 512
#define OUT_CHANNELS 1024
#define NUM_POINTS 17
#define NPAD 19      // pre-wrapped table: j in [0,18] covers raw index [-8, 10]
#define BLOCK 256

typedef int v4i __attribute__((vector_size(4 * sizeof(int))));

#if defined(__gfx1250__) && __has_builtin(__builtin_amdgcn_tanhf)
#define FAST_TANH(x) __builtin_amdgcn_tanhf(x)
#elif defined(__gfx1250__) && __has_builtin(__builtin_amdgcn_tanh_f32)
#define FAST_TANH(x) __builtin_amdgcn_tanh_f32(x)
#else
#define FAST_TANH(x) tanhf(x)
#endif

#if __has_builtin(__builtin_amdgcn_sinf) && __has_builtin(__builtin_amdgcn_cosf)
#define HW_SIN(x) __builtin_amdgcn_sinf(x)   // computes sin(2*pi*x)
#define HW_COS(x) __builtin_amdgcn_cosf(x)
#define ANG_SCALE 0.15915494309189535f       // 1/(2*pi)
#else
#define HW_SIN(x) __sinf(x)
#define HW_COS(x) __cosf(x)
#define ANG_SCALE 1.0f
#endif

__global__ __launch_bounds__(BLOCK) void macunit_kernel(
    const float* __restrict__ data,
    const float* __restrict__ angles,
    const float* __restrict__ velocity,
    const float* __restrict__ attention,
    const float* __restrict__ coeff_p,
    const float* __restrict__ bias_p,
    float* __restrict__ out,
    int n_groups)  // useful threads = batch * 128
{
  // s_pair[j] = (velocity[(j+9)%17], angles[(j+9)%17]): table pre-wrapped so the
  // biased index j = floor(raw_index + 8) in [0,17] needs no negative-wrap selects.
  __shared__ float2 s_pair[NPAD];
  __shared__ float4 s_att[OUT_CHANNELS / 4];   // 4 KB attention table

  const int t = threadIdx.x;

#if defined(__gfx1250__) && \
    __has_builtin(__builtin_amdgcn_global_load_async_to_lds_b128) && \
    __has_builtin(__builtin_amdgcn_global_load_async_to_lds_b32)
  // CDNA5 async DMA staging: attention (4 KB, one b128 per lane) and the
  // interleaved pre-wrapped (velo, ang) table. Args: (global src, LDS dst, off, cpol).
  __builtin_amdgcn_global_load_async_to_lds_b128(
      (v4i*)(attention + 4 * t), (v4i*)(&s_att[t]), 0, 0);
  if (t < 2 * NPAD) {
    const int j = (t < NPAD) ? t : t - NPAD;
    const int m = (j + 9) % NUM_POINTS;
    const float* src = (t < NPAD) ? (velocity + m) : (angles + m);
    float*       dst = (t < NPAD) ? (&s_pair[j].x) : (&s_pair[j].y);
    __builtin_amdgcn_global_load_async_to_lds_b32((int*)src, (int*)dst, 0, 0);
  }
#if __has_builtin(__builtin_amdgcn_s_wait_asynccnt)
  __builtin_amdgcn_s_wait_asynccnt(0);
#else
  asm volatile("s_wait_asynccnt 0" ::: "memory");
#endif
#else
  // Fallback staging path (also used by the host pass).
  s_att[t] = ((const float4*)attention)[t];
  if (t < NPAD) {
    const int m = (t + 9) % NUM_POINTS;
    s_pair[t].x = velocity[m];
    s_pair[t].y = angles[m];
  }
#endif
  __syncthreads();

  const int gid = blockIdx.x * BLOCK + t;
  if (gid >= n_groups) return;

  const float coeff = coeff_p[0];
  const float bias  = bias_p[0];

  const int cg = gid & 127;  // channel-group within the row (4 input channels)

  // data4[gid] == data[row][cg*4 .. cg*4+3]  (gid = row*128+cg)
  const float4 d4 = ((const float4*)data)[gid];
  float x[4] = {d4.x, d4.y, d4.z, d4.w};

#pragma unroll
  for (int s = 0; s < 3; ++s) {
#pragma unroll
    for (int k = 0; k < 4; ++k) {
      float xv = x[k];
      // raw index = 1 + tanh(x*coeff+bias)*8.5 in (-7.5, 9.5); bias by +8 so
      // j = floor(idx) in [0,17], j+1 in [1,18] index the pre-wrapped table.
      float th  = FAST_TANH(fmaf(xv, coeff, bias));
      float idx = fmaf(th, 8.5f, 9.0f);
      float fi  = floorf(idx);
      float pos = idx - fi;
      int   j   = (int)fi;
      float2 p0 = s_pair[j];          // ds_load_b64
      float2 p1 = s_pair[j + 1];      // ds_load_b64 (same base, offset:8)
      float velo = fmaf(pos, p1.x - p0.x, p0.x);
      float ang  = fmaf(pos, p1.y - p0.y, p0.y);
      float an   = ang * ANG_SCALE;
      float sn   = HW_SIN(an);
      float cs   = HW_COS(an);
      // velo*cos + x*velo*sin = velo*(cos + x*sin)
      float r = velo * fmaf(xv, sn, cs);
      x[k] = fmaf(r, 0.3333333333333333f, xv);
    }
  }

  // 8 outputs: oc = 8*cg + 2k + f, weight = attention[oc]
  const float4 a0 = s_att[cg * 2];
  const float4 a1 = s_att[cg * 2 + 1];
  float4 o0, o1;
  o0.x = a0.x * x[0]; o0.y = a0.y * x[0]; o0.z = a0.z * x[1]; o0.w = a0.w * x[1];
  o1.x = a1.x * x[2]; o1.y = a1.y * x[2]; o1.z = a1.z * x[3]; o1.w = a1.w * x[3];

  float4* out4 = (float4*)out;
  out4[gid * 2]     = o0;   // out[row*1024 + 8*cg + 0..3]
  out4[gid * 2 + 1] = o1;   // out[row*1024 + 8*cg + 4..7]
}

extern "C" void kernel_launch(void* const* d_in, const int* in_sizes, int n_in,
                              void* d_out, int out_size, void* d_ws, size_t ws_size,
                              hipStream_t stream) {
  const float* data      = (const float*)d_in[0];
  const float* angles    = (const float*)d_in[1];
  const float* velocity  = (const float*)d_in[2];
  const float* attention = (const float*)d_in[3];
  const float* coeff     = (const float*)d_in[4];
  const float* bias      = (const float*)d_in[5];
  float* out = (float*)d_out;

  const int batch    = in_sizes[0] / IN_CHANNELS;       // 32768
  const int n_groups = batch * (IN_CHANNELS / 4);       // one thread per 4 input channels
  const int grid     = (n_groups + BLOCK - 1) / BLOCK;  // 16384 blocks of 256 (8 wave32)

  macunit_kernel<<<grid, BLOCK, 0, stream>>>(data, angles, velocity, attention,
                                             coeff, bias, out, n_groups);
}